// GTF_35115652612104
// MI455X (gfx1250) — compile-verified
//
#include <hip/hip_runtime.h>
#include <math.h>

typedef float v2f __attribute__((ext_vector_type(2)));
typedef float v8f __attribute__((ext_vector_type(8)));

#define DOUT 64

// ---------------------------------------------------------------------------
// Dense GEMM: Y[N x 64] = X[N x din] @ W[din x 64] + bias, fp32 WMMA 16x16x4.
// One wave per 16x16 output tile. N % 16 == 0 (50000 = 3125*16), EXEC all-1s.
// A-frag (16x4 f32): lane L -> row = L%16; VGPR0/1 = K + {0,1} (lanes 0-15)
//                    or K + {2,3} (lanes 16-31).
// B-frag (4x16 f32): VGPR j, lane L -> K-row = j + 2*(L/16), col = L%16.
// C/D (16x16 f32):   VGPR j, lane L -> row = j + 8*(L/16),  col = L%16.
// ---------------------------------------------------------------------------
__global__ void gemm16_wmma_f32(const float* __restrict__ X,
                                const float* __restrict__ W,
                                const float* __restrict__ bias,
                                float* __restrict__ Y,
                                int din)
{
    const int lane  = threadIdx.x & 31;
    const int wave  = threadIdx.x >> 5;       // 0..3 -> N tile
    const int mt    = blockIdx.x;             // M tile (16 rows)
    const int r     = lane & 15;
    const int khalf = (lane >> 4) << 1;       // 0 or 2
    const int col   = wave * 16 + r;

    const float* __restrict__ xrow = X + (size_t)(mt * 16 + r) * din;

    v8f acc = (v8f){0.f, 0.f, 0.f, 0.f, 0.f, 0.f, 0.f, 0.f};
    for (int k0 = 0; k0 < din; k0 += 4) {
        v2f a;
        a.x = xrow[k0 + khalf];
        a.y = xrow[k0 + khalf + 1];
        v2f b;
        b.x = W[(size_t)(k0 + khalf) * DOUT + col];
        b.y = W[(size_t)(k0 + khalf + 1) * DOUT + col];
        acc = __builtin_amdgcn_wmma_f32_16x16x4_f32(
            /*neg_a=*/false, a, /*neg_b=*/false, b,
            /*c_mod=*/(short)0, acc, /*reuse_a=*/false, /*reuse_b=*/false);
    }

    const float bb = bias[col];
    const int nh = lane >> 4;
#pragma unroll
    for (int j = 0; j < 8; ++j) {
        const int m = j + 8 * nh;
        Y[(size_t)(mt * 16 + m) * DOUT + col] = acc[j] + bb;
    }
}

// ---------------------------------------------------------------------------
// Ordered-int encoding of float for atomicMax (handles negatives correctly)
// ---------------------------------------------------------------------------
__device__ __forceinline__ int f32_ord(float f)
{
    int i = __float_as_int(f);
    return (i >= 0) ? i : (i ^ 0x7FFFFFFF);
}
__device__ __forceinline__ float ord_f32(int i)
{
    return __int_as_float((i >= 0) ? i : (i ^ 0x7FFFFFFF));
}

__global__ void zero_f32(float* __restrict__ p, int n)
{
    int i = blockIdx.x * blockDim.x + threadIdx.x;
    if (i < n) p[i] = 0.f;
}

__global__ void init_node(float* __restrict__ denom, int* __restrict__ mmax, int n)
{
    int i = blockIdx.x * blockDim.x + threadIdx.x;
    if (i < n) {
        denom[i] = 0.f;
        mmax[i]  = (int)0x80000000;   // < ordered(-inf)
    }
}

// score[e] = dot(q[dst], k[src]) / sqrt(64);  segment max via ordered atomicMax
__global__ void edge_scores(const float* __restrict__ q, const float* __restrict__ k,
                            const int* __restrict__ src, const int* __restrict__ dst,
                            float* __restrict__ score, int* __restrict__ mmax, int E)
{
    int e = blockIdx.x * blockDim.x + threadIdx.x;
    if (e >= E) return;
    const int sN = src[e], dN = dst[e];
    const float4* __restrict__ qa = (const float4*)(q + (size_t)dN * DOUT);
    const float4* __restrict__ ka = (const float4*)(k + (size_t)sN * DOUT);
    float acc = 0.f;
#pragma unroll
    for (int i = 0; i < DOUT / 4; ++i) {
        float4 a = qa[i];
        float4 b = ka[i];
        acc += a.x * b.x + a.y * b.y + a.z * b.z + a.w * b.w;
    }
    acc *= 0.125f;   // 1/sqrt(64)
    score[e] = acc;
    atomicMax(&mmax[dN], f32_ord(acc));
}

// score[e] <- exp(score[e] - max[dst]);  denom[dst] += score[e]
__global__ void edge_expsum(float* __restrict__ score,
                            const int* __restrict__ dst,
                            const int* __restrict__ mmax,
                            float* __restrict__ denom, int E)
{
    int e = blockIdx.x * blockDim.x + threadIdx.x;
    if (e >= E) return;
    const int dN = dst[e];
    const float m  = ord_f32(mmax[dN]);
    const float ev = expf(score[e] - m);
    score[e] = ev;
    atomicAdd(&denom[dN], ev);
}

// agg[dst] += (score[e]/denom[dst]) * v[src]
__global__ void edge_scatter(const float* __restrict__ score,
                             const float* __restrict__ v,
                             const int* __restrict__ src, const int* __restrict__ dst,
                             const float* __restrict__ denom,
                             float* __restrict__ agg, int E)
{
    int e = blockIdx.x * blockDim.x + threadIdx.x;
    if (e >= E) return;
    const int sN = src[e], dN = dst[e];
    const float den   = denom[dN];
    const float alpha = score[e] / (den == 0.f ? 1.f : den);
    const float4* __restrict__ vv = (const float4*)(v + (size_t)sN * DOUT);
    float* __restrict__ out = agg + (size_t)dN * DOUT;
#pragma unroll
    for (int i = 0; i < DOUT / 4; ++i) {
        float4 t = vv[i];
        atomicAdd(out + 4 * i + 0, alpha * t.x);
        atomicAdd(out + 4 * i + 1, alpha * t.y);
        atomicAdd(out + 4 * i + 2, alpha * t.z);
        atomicAdd(out + 4 * i + 3, alpha * t.w);
    }
}

// h = (agg + skip);  optional ELU (in place on agg buffer)
__global__ void epilogue(float* __restrict__ agg, const float* __restrict__ skip,
                         int n, int do_elu)
{
    int i = blockIdx.x * blockDim.x + threadIdx.x;
    if (i >= n) return;
    float val = agg[i] + skip[i];
    if (do_elu) val = (val > 0.f) ? val : expm1f(val);
    agg[i] = val;
}

// ---------------------------------------------------------------------------
// Host orchestration (graph-capture safe: only kernel launches on `stream`)
// ---------------------------------------------------------------------------
extern "C" void kernel_launch(void* const* d_in, const int* in_sizes, int n_in,
                              void* d_out, int out_size, void* d_ws, size_t ws_size,
                              hipStream_t stream)
{
    const float* x  = (const float*)d_in[0];      // [N,128]
    const int*   ei = (const int*)d_in[1];        // [2,E] int32
    const int N = in_sizes[0] / 128;
    const int E = in_sizes[1] / 2;
    const int* src = ei;
    const int* dst = ei + E;

    // workspace layout
    char* ws = (char*)d_ws;
    const size_t nd = (size_t)N * DOUT;
    float* qb    = (float*)ws; ws += nd * sizeof(float);
    float* kb    = (float*)ws; ws += nd * sizeof(float);
    float* vb    = (float*)ws; ws += nd * sizeof(float);
    float* sb    = (float*)ws; ws += nd * sizeof(float);
    float* h     = (float*)ws; ws += nd * sizeof(float);
    float* score = (float*)ws; ws += (size_t)E * sizeof(float);
    float* denom = (float*)ws; ws += (size_t)N * sizeof(float);
    int*   mmax  = (int*)ws;   ws += (size_t)N * sizeof(int);

    const int TPB     = 256;
    const int mtiles  = N / 16;                       // 3125 (N divisible by 16)
    const int eBlocks = (E + TPB - 1) / TPB;
    const int nBlocks = (N + TPB - 1) / TPB;
    const int ndBlk   = (int)((nd + TPB - 1) / TPB);

    const float* xin = x;
    for (int l = 0; l < 3; ++l) {
        const int din = (l == 0) ? 128 : DOUT;
        const float* Wq = (const float*)d_in[2 + l * 8 + 0];
        const float* bq = (const float*)d_in[3 + l * 8 + 0];
        const float* Wk = (const float*)d_in[2 + l * 8 + 2];
        const float* bk = (const float*)d_in[3 + l * 8 + 2];
        const float* Wv = (const float*)d_in[2 + l * 8 + 4];
        const float* bv = (const float*)d_in[3 + l * 8 + 4];
        const float* Ws = (const float*)d_in[2 + l * 8 + 6];
        const float* bs = (const float*)d_in[3 + l * 8 + 6];

        gemm16_wmma_f32<<<mtiles, 128, 0, stream>>>(xin, Wq, bq, qb, din);
        gemm16_wmma_f32<<<mtiles, 128, 0, stream>>>(xin, Wk, bk, kb, din);
        gemm16_wmma_f32<<<mtiles, 128, 0, stream>>>(xin, Wv, bv, vb, din);
        gemm16_wmma_f32<<<mtiles, 128, 0, stream>>>(xin, Ws, bs, sb, din);

        float* agg = (l == 2) ? (float*)d_out : h;
        zero_f32 <<<ndBlk,  TPB, 0, stream>>>(agg, (int)nd);
        init_node<<<nBlocks, TPB, 0, stream>>>(denom, mmax, N);

        edge_scores <<<eBlocks, TPB, 0, stream>>>(qb, kb, src, dst, score, mmax, E);
        edge_expsum <<<eBlocks, TPB, 0, stream>>>(score, dst, mmax, denom, E);
        edge_scatter<<<eBlocks, TPB, 0, stream>>>(score, vb, src, dst, denom, agg, E);

        epilogue<<<ndBlk, TPB, 0, stream>>>(agg, sb, (int)nd, (l < 2) ? 1 : 0);

        xin = h;  // next layer input (safe: next GEMMs read h before it is re-zeroed)
    }
}